// PLIF_56659208569476
// MI455X (gfx1250) — compile-verified
//
#include <hip/hip_runtime.h>
#include <stdint.h>

// PLIF forward scan, T=16, N = B*C*H*W elements per timestep.
// Memory-bound streaming scan: stage x[t] tiles into LDS with CDNA5 async
// global->LDS copies (ASYNCcnt), 4-deep ring pipeline, NT load/store policy.

typedef float v4f __attribute__((ext_vector_type(4)));

#define T_STEPS 16
#define BLOCK   256
#define DEPTH   4                       // async pipeline depth (ring slots)
#define SLOT_BYTES (BLOCK * 16)         // one v4f per thread per slot

__device__ __forceinline__ void async_load_b128(uint32_t lds_addr, uint64_t gaddr) {
  // GLOBAL_LOAD_ASYNC_TO_LDS_B128: VDST = per-lane LDS byte address,
  // VADDR = per-lane 64-bit global address, no SADDR. Non-temporal: the
  // 268 MB input stream is read exactly once (bigger than the 192 MB L2).
  asm volatile("global_load_async_to_lds_b128 %0, %1, off th:TH_LOAD_NT"
               :
               : "v"(lds_addr), "v"(gaddr)
               : "memory");
}

__device__ __forceinline__ void wait_asynccnt_le(int n) {
  // n is a constant after full unroll; dead branches fold away.
  switch (n) {
    case 0:  asm volatile("s_wait_asynccnt 0x0" ::: "memory"); break;
    case 1:  asm volatile("s_wait_asynccnt 0x1" ::: "memory"); break;
    case 2:  asm volatile("s_wait_asynccnt 0x2" ::: "memory"); break;
    default: asm volatile("s_wait_asynccnt 0x3" ::: "memory"); break;
  }
}

__global__ __launch_bounds__(BLOCK)
void plif_scan_kernel(const float* __restrict__ x, float* __restrict__ out, int n4) {
  __shared__ v4f buf[DEPTH][BLOCK];

  const int tid = (int)threadIdx.x;
  const long long idx4 = (long long)blockIdx.x * BLOCK + tid;   // float4 index
  if (idx4 >= n4) return;

  const uint64_t xin     = (uint64_t)(uintptr_t)x + (uint64_t)idx4 * 16u;
  const uint64_t tstride = (uint64_t)n4 * 16u;                  // bytes between timesteps
  float* o = out + idx4 * 4;

  // Low 32 bits of the generic address of a __shared__ object are the LDS
  // byte offset (aperture hi bits live in [63:32]).
  const uint32_t lds0 = (uint32_t)(uintptr_t)(&buf[0][tid]);

  // Prime the pipeline: timesteps 0..DEPTH-2 in flight.
#pragma unroll
  for (int t = 0; t < DEPTH - 1; ++t)
    async_load_b128(lds0 + (uint32_t)t * SLOT_BYTES, xin + (uint64_t)t * tstride);

  v4f v = {0.f, 0.f, 0.f, 0.f};

#pragma unroll
  for (int t = 0; t < T_STEPS; ++t) {
    // Keep DEPTH-1 timesteps prefetched ahead.
    if (t + DEPTH - 1 < T_STEPS)
      async_load_b128(lds0 + (uint32_t)((t + DEPTH - 1) % DEPTH) * SLOT_BYTES,
                      xin + (uint64_t)(t + DEPTH - 1) * tstride);

    // Async loads complete in order: waiting ASYNCcnt <= (in-flight minus the
    // one we need) guarantees timestep t's slot is resident.
    int pend = T_STEPS - 1 - t;
    if (pend > DEPTH - 1) pend = DEPTH - 1;
    wait_asynccnt_le(pend);

    v4f xt = buf[t % DEPTH][tid];       // ds_load_b128 (dscnt-waited before use)

    // charge / fire / soft reset (v_th == 1.0, so spike*v_th == spike)
    v += xt;
    v4f sp;
    sp.x = v.x > 1.0f ? 1.0f : 0.0f;
    sp.y = v.y > 1.0f ? 1.0f : 0.0f;
    sp.z = v.z > 1.0f ? 1.0f : 0.0f;
    sp.w = v.w > 1.0f ? 1.0f : 0.0f;
    v -= sp;

    // Output stream is written once and never re-read: NT store (b128).
    __builtin_nontemporal_store(sp, (v4f*)(o + (long long)t * n4 * 4));
  }
}

extern "C" void kernel_launch(void* const* d_in, const int* in_sizes, int n_in,
                              void* d_out, int out_size, void* d_ws, size_t ws_size,
                              hipStream_t stream) {
  (void)n_in; (void)out_size; (void)d_ws; (void)ws_size;

  const float* x = (const float*)d_in[0];
  float* out = (float*)d_out;

  const long long total = (long long)in_sizes[0];       // T * B*C*H*W
  const int n4 = (int)(total / T_STEPS / 4);            // float4 per timestep
  const int blocks = (n4 + BLOCK - 1) / BLOCK;

  plif_scan_kernel<<<blocks, BLOCK, 0, stream>>>(x, out, n4);
}